// GraphNormDGL_49134425866999
// MI455X (gfx1250) — compile-verified
//
#include <hip/hip_runtime.h>

// GraphNorm (DGL-style) for MI455X / gfx1250.
//
// Roofline: 512MB in + 512MB out @ 23.3 TB/s => ~45us floor; memory bound.
// Strategy: read HBM exactly once. Each workgroup owns one (graph, 64-col)
// tile = 512x64 fp32 = 128KB, staged into LDS by ONE Tensor-Data-Mover
// descriptor (tensor_load_to_lds, waited with s_wait_tensorcnt). Column
// sums / sums-of-squares are computed with V_WMMA_F32_16X16X4_F32 using a
// ones A-matrix (C[m][n] = sum_k B[k][n] for all m). Variance uses the
// moment identity var = E[x^2] + m^2*s*(s-2) so one stats pass suffices.
// Final pass: out = x*scale + shift (per-column FMA), coalesced b128 stores.

#define HID      256
#define TCOLS    64            // feature columns per workgroup tile
#define NTHREADS 1024          // 32 wave32 waves
#define EPS      1e-6f

typedef float v2f  __attribute__((ext_vector_type(2)));
typedef float v8f  __attribute__((ext_vector_type(8)));
typedef unsigned int u32x4 __attribute__((ext_vector_type(4)));
typedef int   i32x4 __attribute__((ext_vector_type(4)));
typedef int   i32x8 __attribute__((ext_vector_type(8)));

__global__ __launch_bounds__(NTHREADS)
void graphnorm_k(const float* __restrict__ x,
                 const int*   __restrict__ batch_list,
                 const float* __restrict__ weight,
                 const float* __restrict__ bias,
                 const float* __restrict__ mscale,
                 float*       __restrict__ out,
                 int rows)                      // nodes per graph (512)
{
    extern __shared__ float smem[];
    float* tile  = smem;                              // rows*TCOLS floats
    float* sumX  = smem + (size_t)rows * TCOLS;       // TCOLS
    float* sumX2 = sumX  + TCOLS;                     // TCOLS
    float* scA   = sumX2 + TCOLS;                     // TCOLS (weight*rstd)
    float* shA   = scA   + TCOLS;                     // TCOLS (bias - m*s*scale)

    const int tid = threadIdx.x;
    const int g   = blockIdx.y;        // graph
    const int h   = blockIdx.x;        // 64-column tile index (0..3)
    const size_t gstart = (size_t)g * rows;           // first node of graph
    const float* __restrict__ gsrc = x   + gstart * HID + h * TCOLS;
    float*       __restrict__ gdst = out + gstart * HID + h * TCOLS;

    if (tid < TCOLS) { sumX[tid] = 0.0f; sumX2[tid] = 0.0f; }

    // ---------------- Stage tile (rows x 64 fp32) into LDS ----------------
#if defined(__has_builtin) && __has_builtin(__builtin_amdgcn_tensor_load_to_lds) && __has_builtin(__builtin_amdgcn_s_wait_tensorcnt)
    if (tid == 0) {
        // Tensor DMA descriptor (D#), 2D tile: dim0 = 64 elems (row chunk),
        // dim1 = rows, row stride = HID elems, data_size = 4B.
        unsigned long long ga = (unsigned long long)(size_t)gsrc;
        u32x4 g0;
        g0[0] = 1u;                                  // count=1 (valid), no gather
        g0[1] = 0u;                                  // lds_addr: tile at LDS base
        g0[2] = (unsigned)ga;                        // global_addr[31:0]
        g0[3] = (unsigned)((ga >> 32) & 0x01FFFFFFu) // global_addr[56:32]
              | (2u << 30);                          // type = 2 ("image")
        i32x8 g1;
        g1[0] = (int)(2u << 16);                     // data_size = 2 (4 bytes)
        g1[1] = (int)((unsigned)TCOLS << 16);        // tensor_dim0 lo16
        g1[2] = (int)((unsigned)rows  << 16);        // dim0 hi16=0 | tensor_dim1 lo16
        g1[3] = (int)((unsigned)TCOLS << 16);        // dim1 hi16=0 | tile_dim0
        g1[4] = rows;                                // tile_dim1 | tile_dim2=0
        g1[5] = HID;                                 // tensor_dim0_stride lo32
        g1[6] = 0;                                   // stride0 hi | stride1 lo
        g1[7] = 0;                                   // stride1 hi
        i32x4 gz4 = {0, 0, 0, 0};                    // groups 2/3 unused (2D)
        i32x8 gz8 = {0, 0, 0, 0, 0, 0, 0, 0};        // extra operand (clang-23 form)
        // 6-arg form (amdgpu-toolchain / clang-23 HIP headers)
        __builtin_amdgcn_tensor_load_to_lds(g0, g1, gz4, gz4, gz8, 0);
        __builtin_amdgcn_s_wait_tensorcnt(0);
    }
#else
    {   // Fallback: coalesced synchronous b128 loads.
        const int total = rows * (TCOLS / 4);
        for (int q = tid; q < total; q += NTHREADS) {
            int r  = q >> 4;
            int c4 = (q & 15) * 4;
            *(float4*)(tile + r * TCOLS + c4) =
                *(const float4*)(gsrc + (size_t)r * HID + c4);
        }
    }
#endif
    __syncthreads();

    // ------- Column sums & sums-of-squares via WMMA f32 16x16x4 -----------
    // A = ones(16x4)  =>  C[m][n] += sum_k B[k][n]  (identical for every m,
    // and summation is permutation-invariant so B's K<->row mapping is free).
    const int lane = tid & 31;
    const int wave = tid >> 5;
    const int grp  = wave & 3;            // 4 groups of 16 columns
    const int wig  = wave >> 2;           // 8 waves per group
    const int colb = grp * 16;
    const int n    = lane & 15;
    const int kh   = lane >> 4;           // which K-half this lane feeds
    const int rowsPerWave = rows >> 3;    // 64

#if defined(__has_builtin) && __has_builtin(__builtin_amdgcn_wmma_f32_16x16x4_f32)
    {
        v8f acc  = {0,0,0,0,0,0,0,0};
        v8f acc2 = {0,0,0,0,0,0,0,0};
        v2f ones = {1.0f, 1.0f};
        const float* bp = tile + (wig * rowsPerWave + 2 * kh) * TCOLS + colb + n;
        for (int it = 0; it < rowsPerWave; it += 4) {
            float b0 = bp[0];
            float b1 = bp[TCOLS];
            v2f bv  = {b0, b1};
            v2f bv2 = {b0 * b0, b1 * b1};
            acc  = __builtin_amdgcn_wmma_f32_16x16x4_f32(
                       false, ones, false, bv,  (short)0, acc,  false, false);
            acc2 = __builtin_amdgcn_wmma_f32_16x16x4_f32(
                       false, ones, false, bv2, (short)0, acc2, false, false);
            bp += 4 * TCOLS;
        }
        // C VGPR0: lanes 0-15 hold (M=0, N=lane); all M rows identical.
        if (lane < 16) {
            atomicAdd(&sumX [colb + n], acc[0]);     // ds_add_f32
            atomicAdd(&sumX2[colb + n], acc2[0]);
        }
    }
#else
    {   // VALU fallback: 16 threads per column, 32-row strips.
        const int c = tid & (TCOLS - 1);
        const int r0 = (tid >> 6) * (rows >> 4);
        float s = 0.0f, s2 = 0.0f;
        for (int r = r0; r < r0 + (rows >> 4); ++r) {
            float v = tile[r * TCOLS + c];
            s += v; s2 += v * v;
        }
        atomicAdd(&sumX[c], s);
        atomicAdd(&sumX2[c], s2);
    }
#endif
    __syncthreads();

    // ---------------- Per-column normalization constants ------------------
    if (tid < TCOLS) {
        const int gc = h * TCOLS + tid;
        const float cnt = (float)batch_list[g];
        const float inv = 1.0f / cnt;
        const float m   = sumX[tid]  * inv;
        const float e2  = sumX2[tid] * inv;
        const float s   = mscale[gc];
        // E[(x - m*s)^2] = E[x^2] - 2*s*m^2 + s^2*m^2
        const float var  = e2 + m * m * s * (s - 2.0f);
        const float rstd = rsqrtf(var + EPS);
        const float sc   = weight[gc] * rstd;
        scA[tid] = sc;
        shA[tid] = bias[gc] - m * s * sc;            // out = x*sc + sh
    }
    __syncthreads();

    // ---------------- Normalize + store (coalesced b128) ------------------
    {
        const int c4 = (tid & 15) * 4;               // fixed 4 columns/thread
        const float4 sc4 = *(const float4*)(scA + c4);
        const float4 sh4 = *(const float4*)(shA + c4);
        const int iters = rows >> 6;                 // rows*16 chunks / 1024
        int q = tid;
        for (int i = 0; i < iters; ++i, q += NTHREADS) {
            const int r = q >> 4;
            float4 v = *(const float4*)(tile + r * TCOLS + c4);
            float4 o;
            o.x = v.x * sc4.x + sh4.x;
            o.y = v.y * sc4.y + sh4.y;
            o.z = v.z * sc4.z + sh4.z;
            o.w = v.w * sc4.w + sh4.w;
            *(float4*)(gdst + (size_t)r * HID + c4) = o;
        }
    }
}

extern "C" void kernel_launch(void* const* d_in, const int* in_sizes, int n_in,
                              void* d_out, int out_size, void* d_ws, size_t ws_size,
                              hipStream_t stream) {
    const float* x     = (const float*)d_in[0];
    const int*   batch = (const int*)  d_in[1];
    const float* w     = (const float*)d_in[2];
    const float* b     = (const float*)d_in[3];
    const float* ms    = (const float*)d_in[4];
    float*       out   = (float*)d_out;

    const int N    = in_sizes[0] / HID;   // total nodes (524288)
    const int B    = in_sizes[1];         // graphs (1024)
    const int rows = N / B;               // nodes per graph (512)

    dim3 grid(HID / TCOLS, B);            // 4 x 1024 workgroups
    const size_t shmem = (size_t)rows * TCOLS * sizeof(float)
                       + 4 * TCOLS * sizeof(float);   // 128KB tile + stats

    graphnorm_k<<<grid, dim3(NTHREADS), shmem, stream>>>(
        x, batch, w, b, ms, out, rows);
}